// PPRSweep_56676388438732
// MI455X (gfx1250) — compile-verified
//
#include <hip/hip_runtime.h>
#include <stdint.h>

#define N_NODES   100000
#define N_EDGES   1600000
#define D_FEAT    32
#define N_ITERS   10
#define WAVES_PB  8            // 256 threads

// CSR-gather path
#define NPB       64           // nodes per block
#define CAP       2048         // LDS edge-record capacity (expected ~1025 per block)
#define SCAN_B    1024         // elements per scan block

// SoA atomic fallback path
#define TILE2     2048
#define EPW2      (TILE2 / WAVES_PB)

// ===========================================================================
// CDNA5 primitives
// ===========================================================================
__device__ __forceinline__ void async_copy_b128(uint32_t lds_byte_off, uint64_t gaddr) {
    asm volatile("global_load_async_to_lds_b128 %0, %1, off"
                 :: "v"(lds_byte_off), "v"(gaddr) : "memory");
}
__device__ __forceinline__ void wait_async_le0() {
    asm volatile("s_wait_asynccnt 0x0" ::: "memory");
}
__device__ __forceinline__ uint32_t lds_off(const void* p) {   // low 32b = DS offset
    return (uint32_t)(uintptr_t)p;
}
__device__ __forceinline__ void atomic_add_f32_dev(float* p, float v) {
    asm volatile("global_atomic_add_f32 %0, %1, off scope:SCOPE_DEV"
                 :: "v"((uint64_t)(uintptr_t)p), "v"(v) : "memory");
}

// ===========================================================================
// Elementwise helpers
// ===========================================================================
__global__ void fill_kernel(float* __restrict__ p, float v, int n4) {
    int i = blockIdx.x * blockDim.x + threadIdx.x;
    if (i < n4) ((float4*)p)[i] = make_float4(v, v, v, v);
}

__global__ void finalize_kernel(const float* __restrict__ f, const float* __restrict__ h,
                                float* __restrict__ out, int n4) {
    int i = blockIdx.x * blockDim.x + threadIdx.x;
    if (i < n4) {
        float4 a = ((const float4*)f)[i];
        float4 b = ((const float4*)h)[i];
        float4 r;
        r.x = a.x / b.x; r.y = a.y / b.y; r.z = a.z / b.z; r.w = a.w / b.w;
        ((float4*)out)[i] = r;
    }
}

// ===========================================================================
// CSR build: degree histogram -> exclusive scan -> record scatter
// ===========================================================================
__global__ void hist_kernel(const int* __restrict__ edst, int* __restrict__ deg, int n) {
    int e = blockIdx.x * blockDim.x + threadIdx.x;
    if (e < n) atomicAdd(&deg[edst[e]], 1);           // global_atomic_add_u32
}

__global__ void scan1_kernel(const int* __restrict__ deg, int* __restrict__ rs,
                             int* __restrict__ bsums, int n) {
    __shared__ int s[256];
    const int tid  = threadIdx.x;
    const int base = blockIdx.x * SCAN_B + tid * 4;
    int v0 = 0, v1 = 0, v2 = 0, v3 = 0;
    if (base + 0 < n) v0 = deg[base + 0];
    if (base + 1 < n) v1 = deg[base + 1];
    if (base + 2 < n) v2 = deg[base + 2];
    if (base + 3 < n) v3 = deg[base + 3];
    s[tid] = v0 + v1 + v2 + v3;
    __syncthreads();
    #pragma unroll
    for (int off = 1; off < 256; off <<= 1) {         // Hillis-Steele inclusive
        int x = s[tid];
        int y = (tid >= off) ? s[tid - off] : 0;
        __syncthreads();
        s[tid] = x + y;
        __syncthreads();
    }
    const int excl = (tid > 0) ? s[tid - 1] : 0;
    if (tid == 255) bsums[blockIdx.x] = s[255];
    if (base + 0 < n) rs[base + 0] = excl;
    if (base + 1 < n) rs[base + 1] = excl + v0;
    if (base + 2 < n) rs[base + 2] = excl + v0 + v1;
    if (base + 3 < n) rs[base + 3] = excl + v0 + v1 + v2;
}

__global__ void scan2_kernel(int* __restrict__ bsums, int nb) {
    if (threadIdx.x == 0 && blockIdx.x == 0) {
        int run = 0;
        for (int i = 0; i < nb; ++i) { int t = bsums[i]; bsums[i] = run; run += t; }
    }
}

__global__ void scan3_kernel(int* __restrict__ rs, int* __restrict__ cursor,
                             const int* __restrict__ bsums, int n) {
    int i = blockIdx.x * blockDim.x + threadIdx.x;
    if (i < n) {
        int v = rs[i] + bsums[i / SCAN_B];
        rs[i] = v;
        cursor[i] = v;
    }
    if (i == 0) rs[n] = N_EDGES;
}

// record = {src, 0.9*w}: restart blend folded into the stored weight
__global__ void build_kernel(const int* __restrict__ esrc, const int* __restrict__ edst,
                             const float* __restrict__ ew, int* __restrict__ cursor,
                             int2* __restrict__ recs, int n) {
    int e = blockIdx.x * blockDim.x + threadIdx.x;
    if (e < n) {
        int pos = atomicAdd(&cursor[edst[e]], 1);
        recs[pos] = make_int2(esrc[e], __float_as_int(0.9f * ew[e]));
    }
}

// ===========================================================================
// Gather SpMM sweep: no atomics. One wave owns one node (lane = feature).
// Block's 64 contiguous rows => contiguous record range => async-to-LDS stage,
// then ds_load_b64 broadcast per edge + coalesced 128B gather + v_fmac.
// Hn[n*32+lane] = 0.1 + sum(0.9*w*Hc[src*32+lane])  -- written once, plain store.
// ===========================================================================
__global__ void __launch_bounds__(256)
spmm_gather_kernel(const int* __restrict__ rs, const int2* __restrict__ recs,
                   const float* __restrict__ Hc, float* __restrict__ Hn, int nnodes) {
    __shared__ int2 lrec[CAP];                       // 16 KB
    __shared__ int  srs[NPB + 1];

    const int tid   = threadIdx.x;
    const int first = blockIdx.x * NPB;
    const int lastn = min(NPB, nnodes - first);      // >= 1 by grid construction
    if (tid <= lastn) srs[tid] = rs[first + tid];
    __syncthreads();

    const int  ebeg    = srs[0];
    const int  eend    = srs[lastn];
    const int  abeg    = ebeg & ~1;                  // 16B-aligned record start
    const int  cnt     = eend - abeg;
    const bool use_lds = (cnt <= CAP);

    if (use_lds) {                                   // block-uniform branch
        const int chunks = (cnt + 1) >> 1;           // 16B async chunks
        for (int c = tid; c < chunks; c += 256)
            async_copy_b128(lds_off(&lrec[c * 2]),
                            (uint64_t)(uintptr_t)(recs + abeg + c * 2));
        wait_async_le0();
    }
    __syncthreads();

    const int wave = tid >> 5;
    const int lane = tid & 31;
    for (int k = wave; k < lastn; k += WAVES_PB) {
        const int r0 = srs[k];
        const int r1 = srs[k + 1];
        float acc = 0.1f;                            // restart term
        if (use_lds) {
            #pragma unroll 2
            for (int j = r0; j < r1; ++j) {
                int2 rec = lrec[j - abeg];           // ds_load_b64 broadcast
                acc += __int_as_float(rec.y) * Hc[(size_t)rec.x * D_FEAT + lane];
            }
        } else {                                     // rare overflow fallback
            #pragma unroll 2
            for (int j = r0; j < r1; ++j) {
                int2 rec = recs[j];
                acc += __int_as_float(rec.y) * Hc[(size_t)rec.x * D_FEAT + lane];
            }
        }
        Hn[(size_t)(first + k) * D_FEAT + lane] = acc;   // coalesced 128B store
    }
}

// ===========================================================================
// SoA atomic fallback (small workspace)
// ===========================================================================
__global__ void prep_soa_kernel(const float* __restrict__ w, float* __restrict__ w2, int n4) {
    int i = blockIdx.x * blockDim.x + threadIdx.x;
    if (i < n4) {
        float4 x = ((const float4*)w)[i];
        x.x *= 0.9f; x.y *= 0.9f; x.z *= 0.9f; x.w *= 0.9f;
        ((float4*)w2)[i] = x;
    }
}

__global__ void __launch_bounds__(256)
spmm_soa_kernel(const int* __restrict__ esrc, const int* __restrict__ edst,
                const float* __restrict__ ew2, const float* __restrict__ Hc,
                float* __restrict__ Hn) {
    __shared__ int   s_src[TILE2];
    __shared__ int   s_dst[TILE2];
    __shared__ float s_w[TILE2];

    const int tid       = threadIdx.x;
    const int tile_base = blockIdx.x * TILE2;
    const int chunks    = TILE2 / 4;
    for (int c = tid; c < chunks; c += blockDim.x) {
        int ei = tile_base + c * 4;
        int ce = (ei + 4 <= N_EDGES) ? ei : (N_EDGES - 4);
        async_copy_b128(lds_off(&s_src[c * 4]), (uint64_t)(uintptr_t)(esrc + ce));
        async_copy_b128(lds_off(&s_dst[c * 4]), (uint64_t)(uintptr_t)(edst + ce));
        async_copy_b128(lds_off(&s_w  [c * 4]), (uint64_t)(uintptr_t)(ew2  + ce));
    }
    wait_async_le0();
    __syncthreads();

    const int wave = tid >> 5;
    const int lane = tid & 31;
    const int base = wave * EPW2;
    int cnt = N_EDGES - (tile_base + base);
    if (cnt > EPW2) cnt = EPW2;
    #pragma unroll 4
    for (int i = 0; i < cnt; ++i) {
        const int e = base + i;
        const float h = Hc[(size_t)s_src[e] * D_FEAT + lane];
        atomic_add_f32_dev(&Hn[(size_t)s_dst[e] * D_FEAT + lane], s_w[e] * h);
    }
}

// ===========================================================================
extern "C" void kernel_launch(void* const* d_in, const int* in_sizes, int n_in,
                              void* d_out, int out_size, void* d_ws, size_t ws_size,
                              hipStream_t stream) {
    const float* feat = (const float*)d_in[0];
    const float* ew   = (const float*)d_in[1];
    const int*   esrc = (const int*)d_in[2];
    const int*   edst = (const int*)d_in[3];
    float*       out  = (float*)d_out;

    const size_t NF  = (size_t)N_NODES * D_FEAT;       // 3.2M floats
    const int    T   = 256;
    const int    n4h = (int)(NF / 4);                  // 800000 (exact)
    const int    gfh = (n4h + T - 1) / T;
    const int    ge  = (N_EDGES + T - 1) / T;

    // workspace layout (16B-aligned segments)
    char*  base   = (char*)d_ws;
    float* Ha     = (float*)base;                                   // 12.8 MB
    int2*  recs   = (int2*)(base + NF * 4);                         // 12.8 MB
    int*   rs     = (int*) (base + NF * 4 + (size_t)N_EDGES * 8);   // (N+4)*4
    int*   cursor = rs + (N_NODES + 4);
    int*   deg    = cursor + N_NODES;
    int*   bsums  = deg + N_NODES;
    const int nb_scan = (N_NODES + SCAN_B - 1) / SCAN_B;            // 98
    const size_t need_csr = NF * 4 + (size_t)N_EDGES * 8 +
                            (size_t)(3 * N_NODES + 4 + 128) * 4;    // ~26.8 MB

    if (ws_size >= need_csr) {
        // ---- one-time CSR build (per call; deterministic sequence) ----
        fill_kernel<<<(N_NODES / 4 + T - 1) / T, T, 0, stream>>>((float*)deg, 0.0f, N_NODES / 4);
        hist_kernel<<<ge, T, 0, stream>>>(edst, deg, N_EDGES);
        scan1_kernel<<<nb_scan, T, 0, stream>>>(deg, rs, bsums, N_NODES);
        scan2_kernel<<<1, 64, 0, stream>>>(bsums, nb_scan);
        scan3_kernel<<<(N_NODES + T - 1) / T, T, 0, stream>>>(rs, cursor, bsums, N_NODES);
        build_kernel<<<ge, T, 0, stream>>>(esrc, edst, ew, cursor, recs, N_EDGES);

        // ---- 10 atomic-free gather sweeps, ping-pong Ha <-> d_out ----
        fill_kernel<<<gfh, T, 0, stream>>>(Ha, 1.0f, n4h);          // H0 = 1
        const int gridN = (N_NODES + NPB - 1) / NPB;                // 1563
        float* cur = Ha;
        float* nxt = out;
        for (int it = 0; it < N_ITERS; ++it) {
            spmm_gather_kernel<<<gridN, T, 0, stream>>>(rs, recs, cur, nxt, N_NODES);
            float* tmp = cur; cur = nxt; nxt = tmp;
        }
        finalize_kernel<<<gfh, T, 0, stream>>>(feat, cur, out, n4h);   // cur == Ha
    } else {
        // ---- small-ws fallback: SoA + hardware f32 atomics ----
        float* ew2 = Ha + NF;
        const int n4e    = N_EDGES / 4;
        const int ntiles = (N_EDGES + TILE2 - 1) / TILE2;
        prep_soa_kernel<<<(n4e + T - 1) / T, T, 0, stream>>>(ew, ew2, n4e);
        fill_kernel<<<gfh, T, 0, stream>>>(Ha, 1.0f, n4h);
        float* cur = Ha;
        float* nxt = out;
        for (int it = 0; it < N_ITERS; ++it) {
            fill_kernel<<<gfh, T, 0, stream>>>(nxt, 0.1f, n4h);
            spmm_soa_kernel<<<ntiles, T, 0, stream>>>(esrc, edst, ew2, cur, nxt);
            float* tmp = cur; cur = nxt; nxt = tmp;
        }
        finalize_kernel<<<gfh, T, 0, stream>>>(feat, cur, out, n4h);
    }
}